// PertNet_68487548502050
// MI455X (gfx1250) — compile-verified
//
#include <hip/hip_runtime.h>

#define GG 5000
#define HH 64
#define BB 32
#define NN (GG * BB)       // 160000
#define EE1 100000
#define EE2 200000
#define EPSF 1e-5f
#define LAMB 0.2f

typedef __attribute__((ext_vector_type(2))) float v2f;
typedef __attribute__((ext_vector_type(4))) float v4f;
typedef __attribute__((ext_vector_type(8))) float v8f;

// V_WMMA_F32_16X16X4_F32: D(16x16,f32) = A(16x4,f32) * B(4x16,f32) + C
// A frag: lane&15 = M, a.x = A[M][k0+2*hi], a.y = A[M][k0+2*hi+1]  (hi = lane>>4)
// B frag: lane&15 = N, b.x = B[k0+2*hi][N], b.y = B[k0+2*hi+1][N]
// D: vgpr r -> row = r + hi*8, col = lane&15
static __device__ __forceinline__ v8f wmma4(v2f a, v2f b, v8f c) {
    return __builtin_amdgcn_wmma_f32_16x16x4_f32(false, a, false, b, (short)0, c,
                                                 false, false);
}

static __device__ __forceinline__ float wave_red(float v) {
    for (int off = 16; off > 0; off >>= 1) v += __shfl_xor(v, off, 32);
    return v;
}

// ---------------------------------------------------------------- fill
__global__ void k_fill(float* p, float v, int n) {
    int i = blockIdx.x * blockDim.x + threadIdx.x;
    int st = gridDim.x * blockDim.x;
    for (; i < n; i += st) p[i] = v;
}

// ------------------------------------------------- renorm rows + bn_emb stats
__global__ __launch_bounds__(256) void k_renorm(const float* ge, const float* pe,
                                                const float* po, float* Eg, float* Rp,
                                                float* Pg, float* statsE) {
    int lane = threadIdx.x & 31;
    int wv = (blockIdx.x * blockDim.x + threadIdx.x) >> 5;
    int nw = (gridDim.x * blockDim.x) >> 5;
    float s0 = 0.f, q0 = 0.f, s1 = 0.f, q1 = 0.f;
    for (int g = wv; g < GG; g += nw) {
        // gene embedding (also feeds bn_emb stats; stats over G == stats over N)
        float a = ge[g * 64 + lane], b = ge[g * 64 + lane + 32];
        float n2 = wave_red(a * a + b * b);
        float sc = fminf(1.0f, 1.0f / (sqrtf(n2) + 1e-7f));
        float ea = a * sc, eb = b * sc;
        Eg[g * 64 + lane] = ea;
        Eg[g * 64 + lane + 32] = eb;
        s0 += ea; q0 += ea * ea; s1 += eb; q1 += eb * eb;
        // pert embedding
        a = pe[g * 64 + lane]; b = pe[g * 64 + lane + 32];
        n2 = wave_red(a * a + b * b);
        sc = fminf(1.0f, 1.0f / (sqrtf(n2) + 1e-7f));
        Rp[g * 64 + lane] = a * sc;
        Rp[g * 64 + lane + 32] = b * sc;
        // pos embedding
        a = po[g * 64 + lane]; b = po[g * 64 + lane + 32];
        n2 = wave_red(a * a + b * b);
        sc = fminf(1.0f, 1.0f / (sqrtf(n2) + 1e-7f));
        Pg[g * 64 + lane] = a * sc;
        Pg[g * 64 + lane + 32] = b * sc;
    }
    atomicAdd(&statsE[lane], s0);
    atomicAdd(&statsE[lane + 32], s1);
    atomicAdd(&statsE[64 + lane], q0);
    atomicAdd(&statsE[64 + lane + 32], q1);
}

// ---------------------------------------------------------------- degrees
__global__ void k_deg(const int* ce, const float* cw, const int* se, const float* sw,
                      float* degC, float* degS) {
    int i = blockIdx.x * blockDim.x + threadIdx.x;
    int st = gridDim.x * blockDim.x;
    for (; i < EE1 + EE2; i += st) {
        if (i < EE1) {
            atomicAdd(&degC[ce[EE1 + i]], cw[i]);
        } else {
            int e = i - EE1;
            atomicAdd(&degS[se[EE2 + e]], sw[e]);
        }
    }
}

// -------------------------------- BN finalize: a = g/sqrt(v+eps), c = b - m*a
__global__ void k_finalize(const float* stats, const float* gma, const float* bta,
                           float* a, float* c, int nF, float cnt) {
    int h = blockIdx.x * blockDim.x + threadIdx.x;
    if (h < nF) {
        float m = stats[h] / cnt;
        float v = stats[nF + h] / cnt - m * m;
        float af = gma[h] * rsqrtf(v + EPSF);
        a[h] = af;
        c[h] = bta[h] - m * af;
    }
}

// ---------------------------------------- Tg = relu(bn_emb(Eg)); dinv tables
__global__ void k_gmisc(const float* Eg, const float* aE, const float* cE,
                        const float* degC, const float* degS, float* Tg, float* dinvC,
                        float* dc2, float* dinvS, float* ds2) {
    int i = blockIdx.x * blockDim.x + threadIdx.x;
    int st = gridDim.x * blockDim.x;
    for (; i < GG * 64; i += st) {
        int h = i & 63;
        Tg[i] = fmaxf(aE[h] * Eg[i] + cE[h], 0.0f);
        if (i < GG) {
            dinvC[i] = rsqrtf(degC[i]);
            dc2[i] = 1.0f / degC[i];
            dinvS[i] = rsqrtf(degS[i]);
            ds2[i] = 1.0f / degS[i];
        }
    }
}

// --------------------------- sgconv scatter: AGG[dst] += norm * X[src]
__global__ void k_scatter(const int* ed, const float* w, const float* dinv,
                          const float* X, float* AGG, int E) {
    int i = blockIdx.x * blockDim.x + threadIdx.x;
    int st = gridDim.x * blockDim.x;
    int total = E * 64;
    for (; i < total; i += st) {
        int e = i >> 6, h = i & 63;
        int s = ed[e], d = ed[E + e];
        float nrm = dinv[s] * w[e] * dinv[d];
        atomicAdd(&AGG[d * 64 + h], nrm * X[s * 64 + h]);
    }
}

// --- small WMMA GEMM: OUT(M x 64) = (A1 [+ rs*A2])(M x 64) @ W(64 x 64) + bias
// Templated so the A2-combine and relu are compile-time (no per-K-step branches).
template <bool HASA2, bool RELU>
__global__ __launch_bounds__(256) void k_gemm64(const float* __restrict__ A1,
                                                const float* __restrict__ A2,
                                                const float* __restrict__ rs,
                                                const float* __restrict__ W,
                                                const float* __restrict__ bias,
                                                float* __restrict__ OUT, int M) {
    int lane = threadIdx.x & 31;
    int w = threadIdx.x >> 5;  // 8 waves: 2 row tiles x 4 col tiles
    int nRT = (M + 15) >> 4;
    int rt = blockIdx.x * 2 + (w >> 2);
    if (rt >= nRT) return;  // uniform per wave
    int m = lane & 15, hi = lane >> 4;
    int row = rt * 16 + m;
    int rc = row < M ? row : 0;
    float gate = row < M ? 1.0f : 0.0f;
    float rsv = HASA2 ? rs[rc] : 0.0f;
    int col = (w & 3) * 16 + m;
    const float* a1p = A1 + (size_t)rc * 64;
    const float* a2p = HASA2 ? A2 + (size_t)rc * 64 : a1p;
    v8f acc = {};
#pragma unroll
    for (int k0 = 0; k0 < 64; k0 += 4) {
        int kk = k0 + hi * 2;
        v2f a, b;
        float x0 = a1p[kk], x1 = a1p[kk + 1];
        if (HASA2) {
            x0 += rsv * a2p[kk];
            x1 += rsv * a2p[kk + 1];
        }
        a.x = gate * x0;
        a.y = gate * x1;
        b.x = W[kk * 64 + col];
        b.y = W[(kk + 1) * 64 + col];
        acc = wmma4(a, b, acc);
    }
#pragma unroll
    for (int r = 0; r < 8; r++) {
        int orow = rt * 16 + r + hi * 8;
        if (orow < M) {
            float v = acc[r] + bias[col];
            if (RELU) v = fmaxf(v, 0.0f);
            OUT[(size_t)orow * 64 + col] = v;
        }
    }
}

// ------------------- vb0 = relu(Tg + 0.2 pos0), vbr = relu(Tg + 0.2 posR)
__global__ void k_vb(const float* Tg, const float* pos0, const float* posR, float* vb0,
                     float* vbr) {
    int i = blockIdx.x * blockDim.x + threadIdx.x;
    int st = gridDim.x * blockDim.x;
    for (; i < GG * 64; i += st) {
        float t = Tg[i];
        vb0[i] = fmaxf(t + 0.2f * pos0[i], 0.0f);
        vbr[i] = fmaxf(t + 0.2f * posR[i], 0.0f);
    }
}

__global__ void k_bumpinit(const float* pb, float* bump2) {
    int i = blockIdx.x * blockDim.x + threadIdx.x;
    if (i < BB * 64) bump2[i] = pb[i & 63];
}

// bump2[b,:] += LAMBDA * pert[b,g] * pge2[g,:]   (pert is sparse)
__global__ void k_bump(const float* x, const float* pge2, float* bump2) {
    int i = blockIdx.x * blockDim.x + threadIdx.x;
    int st = gridDim.x * blockDim.x;
    for (; i < NN; i += st) {
        float p = x[2 * i + 1];
        if (p != 0.0f) {
            int b = i / GG, g = i - b * GG;
            float s = LAMB * p;
            for (int h = 0; h < 64; h++)
                atomicAdd(&bump2[b * 64 + h], s * pge2[g * 64 + h]);
        }
    }
}

static __device__ __forceinline__ float t1_val(int i, int h, const float* vb0,
                                               const float* vbr, const float* bump2,
                                               const float* x, float pw) {
    int b = i / GG, g = i - b * GG;
    const float* V = (b == 0) ? vb0 : vbr;
    return V[g * 64 + h] + bump2[b * 64 + h] + x[2 * i + 1] * pw;
}

// ------------------- stats of t1 over N (bn_pbt) — no materialization
__global__ __launch_bounds__(256) void k_stats1(const float* vb0, const float* vbr,
                                                const float* bump2, const float* x,
                                                const float* pW, float* S1) {
    __shared__ float s[128];
    if (threadIdx.x < 128) s[threadIdx.x] = 0.0f;
    __syncthreads();
    int tot = gridDim.x * blockDim.x;  // multiple of 64 -> h fixed per thread
    int idx0 = blockIdx.x * blockDim.x + threadIdx.x;
    int h = idx0 & 63;
    float pw = pW[h];
    float sl = 0.f, ql = 0.f;
    for (int idx = idx0; idx < NN * 64; idx += tot) {
        float t = t1_val(idx >> 6, h, vb0, vbr, bump2, x, pw);
        sl += t;
        ql += t * t;
    }
    atomicAdd(&s[h], sl);
    atomicAdd(&s[64 + h], ql);
    __syncthreads();
    if (threadIdx.x < 128) atomicAdd(&S1[threadIdx.x], s[threadIdx.x]);
}

// ------------------- stats of u = relu(bn1(t1)) over N (bn_pb)
__global__ __launch_bounds__(256) void k_stats2(const float* vb0, const float* vbr,
                                                const float* bump2, const float* x,
                                                const float* pW, const float* a1,
                                                const float* c1, float* S2) {
    __shared__ float s[128];
    if (threadIdx.x < 128) s[threadIdx.x] = 0.0f;
    __syncthreads();
    int tot = gridDim.x * blockDim.x;
    int idx0 = blockIdx.x * blockDim.x + threadIdx.x;
    int h = idx0 & 63;
    float pw = pW[h], a1h = a1[h], c1h = c1[h];
    float sl = 0.f, ql = 0.f;
    for (int idx = idx0; idx < NN * 64; idx += tot) {
        float t = t1_val(idx >> 6, h, vb0, vbr, bump2, x, pw);
        float u = fmaxf(a1h * t + c1h, 0.0f);
        sl += u;
        ql += u * u;
    }
    atomicAdd(&s[h], sl);
    atomicAdd(&s[64 + h], ql);
    __syncthreads();
    if (threadIdx.x < 128) atomicAdd(&S2[threadIdx.x], s[threadIdx.x]);
}

// ---------- GEMM1: h1pre(N x 128) = v(N x 64) @ rec_W1 + b1 ; v recomputed on the fly
// LDS strides: 68 -> half-wave A banks disjoint (4r+{0,1} vs 4r+{2,3});
// 136 -> 2s mod 64 == 16 separates k and k+2 rows of W across half-waves.
#define AS1 68
#define WS1 136
__global__ __launch_bounds__(256) void k_gemm1(
    const float* __restrict__ vb0, const float* __restrict__ vbr,
    const float* __restrict__ bump2, const float* __restrict__ x,
    const float* __restrict__ pW, const float* __restrict__ a1,
    const float* __restrict__ c1, const float* __restrict__ a2,
    const float* __restrict__ c2, const float* __restrict__ W1,
    const float* __restrict__ b1, float* __restrict__ h1pre, float* __restrict__ S3) {
    __shared__ float As[16 * AS1];
    __shared__ float Ws[64 * WS1];
    __shared__ float sS[256];
    int tid = threadIdx.x;
    // vectorized W1 -> LDS (b128 loads + stores; 136*4 and 4h offsets are 16B aligned)
    for (int t = tid; t < (64 * 128) / 4; t += 256) {
        int k = t >> 5, n4 = (t & 31) * 4;
        *(v4f*)&Ws[k * WS1 + n4] = *(const v4f*)&W1[k * 128 + n4];
    }
    sS[tid] = 0.0f;
    __syncthreads();
    int lane = tid & 31, w = tid >> 5;
    int m = lane & 15, hi = lane >> 4;
    int colg = w * 16 + m;  // wave w owns output columns [16w, 16w+16)
    float bcol = b1[colg];
    float lS = 0.f, lQ = 0.f;
    for (int rt = blockIdx.x; rt < NN / 16; rt += gridDim.x) {
        {  // build 16x64 activation tile, 4-wide: one v4 group per thread
            int r = tid >> 4, h4 = (tid & 15) * 4;
            int i = rt * 16 + r;
            int b = i / GG, g = i - b * GG;
            const float* V = (b == 0) ? vb0 : vbr;
            float px = x[2 * i + 1];
            v4f vv = *(const v4f*)(V + g * 64 + h4);
            v4f bu = *(const v4f*)(bump2 + b * 64 + h4);
            v4f pw4 = *(const v4f*)(pW + h4);
            v4f a14 = *(const v4f*)(a1 + h4);
            v4f c14 = *(const v4f*)(c1 + h4);
            v4f a24 = *(const v4f*)(a2 + h4);
            v4f c24 = *(const v4f*)(c2 + h4);
            v4f res;
#pragma unroll
            for (int j = 0; j < 4; j++) {
                float tt = vv[j] + bu[j] + px * pw4[j];
                float u = fmaxf(a14[j] * tt + c14[j], 0.0f);
                res[j] = fmaxf(a24[j] * u + c24[j], 0.0f);
            }
            *(v4f*)&As[r * AS1 + h4] = res;
        }
        __syncthreads();
        v8f acc = {};
#pragma unroll
        for (int k0 = 0; k0 < 64; k0 += 4) {
            int kk = k0 + hi * 2;
            v2f a, b;
            a.x = As[m * AS1 + kk];
            a.y = As[m * AS1 + kk + 1];
            b.x = Ws[kk * WS1 + colg];
            b.y = Ws[(kk + 1) * WS1 + colg];
            acc = wmma4(a, b, acc);
        }
        int rbase = rt * 16 + hi * 8;
#pragma unroll
        for (int r = 0; r < 8; r++) {
            float v = acc[r] + bcol;
            h1pre[(size_t)(rbase + r) * 128 + colg] = v;
            lS += v;
            lQ += v * v;
        }
        __syncthreads();
    }
    atomicAdd(&sS[colg], lS);
    atomicAdd(&sS[128 + colg], lQ);
    __syncthreads();
    atomicAdd(&S3[tid], sS[tid]);
}

// ---------- GEMM2: rpre(N x 64) = relu(bn3(h1pre))(N x 128) @ rec_W2 + b2
#define AS2 132
#define WS2 72
__global__ __launch_bounds__(256) void k_gemm2(const float* __restrict__ h1pre,
                                               const float* __restrict__ a3,
                                               const float* __restrict__ c3,
                                               const float* __restrict__ W2,
                                               const float* __restrict__ b2,
                                               float* __restrict__ rpre,
                                               float* __restrict__ S4) {
    __shared__ float As[32 * AS2];
    __shared__ float Ws[128 * WS2];
    __shared__ float sS[128];
    int tid = threadIdx.x;
    for (int t = tid; t < (128 * 64) / 4; t += 256) {
        int k = t >> 4, n4 = (t & 15) * 4;
        *(v4f*)&Ws[k * WS2 + n4] = *(const v4f*)&W2[k * 64 + n4];
    }
    if (tid < 128) sS[tid] = 0.0f;
    __syncthreads();
    int lane = tid & 31, w = tid >> 5;
    int m = lane & 15, hi = lane >> 4;
    int rtl = w >> 2;             // 2 row tiles per block
    int colg = (w & 3) * 16 + m;  // 4 col tiles
    float bcol = b2[colg];
    float lS = 0.f, lQ = 0.f;
    int rowStride = gridDim.x * 32;
    for (int gidx = blockIdx.x; gidx < NN / 32; gidx += gridDim.x) {
        // 32x128 hmid tile, 4-wide vector loads + prefetch of next grid-stride tile
        for (int t = tid; t < (32 * 128) / 4; t += 256) {
            int r = t >> 5, h4 = (t & 31) * 4;
            int row = gidx * 32 + r;
            v4f hv = *(const v4f*)(h1pre + (size_t)row * 128 + h4);
            v4f a34 = *(const v4f*)(a3 + h4);
            v4f c34 = *(const v4f*)(c3 + h4);
            v4f res;
#pragma unroll
            for (int j = 0; j < 4; j++) res[j] = fmaxf(a34[j] * hv[j] + c34[j], 0.0f);
            *(v4f*)&As[r * AS2 + h4] = res;
            int nrow = row + rowStride;
            if (nrow < NN)
                __builtin_prefetch(h1pre + (size_t)nrow * 128 + h4, 0, 1);
        }
        __syncthreads();
        v8f acc = {};
        int ra = rtl * 16 + m;
#pragma unroll
        for (int k0 = 0; k0 < 128; k0 += 4) {
            int kk = k0 + hi * 2;
            v2f a, b;
            a.x = As[ra * AS2 + kk];
            a.y = As[ra * AS2 + kk + 1];
            b.x = Ws[kk * WS2 + colg];
            b.y = Ws[(kk + 1) * WS2 + colg];
            acc = wmma4(a, b, acc);
        }
        int rbase = gidx * 32 + rtl * 16 + hi * 8;
#pragma unroll
        for (int r = 0; r < 8; r++) {
            float v = acc[r] + bcol;
            rpre[(size_t)(rbase + r) * 64 + colg] = v;
            lS += v;
            lQ += v * v;
        }
        __syncthreads();
    }
    atomicAdd(&sS[colg], lS);
    atomicAdd(&sS[64 + colg], lQ);
    __syncthreads();
    if (tid < 128) atomicAdd(&S4[tid], sS[tid]);
}

// -------- out[i] = dot(bn4(rpre[i,:]), indv_w1[g,:]) + indv_b1[g] + x[i,0]
__global__ __launch_bounds__(256) void k_out(const float* rpre, const float* a4,
                                             const float* c4, const float* iw,
                                             const float* ib, const float* x,
                                             float* out) {
    int lane = threadIdx.x & 31;
    int wv = (blockIdx.x * blockDim.x + threadIdx.x) >> 5;
    int nw = (gridDim.x * blockDim.x) >> 5;
    int h0 = lane, h1 = lane + 32;
    float a0 = a4[h0], cc0 = c4[h0], aa1 = a4[h1], cc1 = c4[h1];
    for (int i = wv; i < NN; i += nw) {
        int g = i - (i / GG) * GG;
        float r0 = a0 * rpre[(size_t)i * 64 + h0] + cc0;
        float r1 = aa1 * rpre[(size_t)i * 64 + h1] + cc1;
        float s = r0 * iw[g * 64 + h0] + r1 * iw[g * 64 + h1];
        s = wave_red(s);
        if (lane == 0) out[i] = s + ib[g] + x[2 * i];
    }
}

extern "C" void kernel_launch(void* const* d_in, const int* in_sizes, int n_in,
                              void* d_out, int out_size, void* d_ws, size_t ws_size,
                              hipStream_t stream) {
    const float* x = (const float*)d_in[0];
    const int* ce = (const int*)d_in[2];
    const float* cw = (const float*)d_in[3];
    const int* se = (const int*)d_in[4];
    const float* sw = (const float*)d_in[5];
    const float* gene_emb = (const float*)d_in[6];
    const float* pert_emb = (const float*)d_in[7];
    const float* emb_pos = (const float*)d_in[8];
    const float* posW = (const float*)d_in[9];
    const float* posb = (const float*)d_in[10];
    const float* simW0 = (const float*)d_in[11];
    const float* simb0 = (const float*)d_in[12];
    const float* simW1 = (const float*)d_in[13];
    const float* simb1 = (const float*)d_in[14];
    const float* pW = (const float*)d_in[15];
    const float* pb = (const float*)d_in[16];
    const float* bnEg = (const float*)d_in[17];
    const float* bnEb = (const float*)d_in[18];
    const float* bn1g = (const float*)d_in[19];
    const float* bn1b = (const float*)d_in[20];
    const float* bn2g = (const float*)d_in[21];
    const float* bn2b = (const float*)d_in[22];
    const float* rW1 = (const float*)d_in[23];
    const float* rb1 = (const float*)d_in[24];
    const float* rbn1g = (const float*)d_in[25];
    const float* rbn1b = (const float*)d_in[26];
    const float* rW2 = (const float*)d_in[27];
    const float* rb2 = (const float*)d_in[28];
    const float* rbn2g = (const float*)d_in[29];
    const float* rbn2b = (const float*)d_in[30];
    const float* iw = (const float*)d_in[31];
    const float* ib = (const float*)d_in[32];
    float* out = (float*)d_out;

    float* ws = (float*)d_ws;
    size_t o = 0;
    auto alloc = [&](size_t n) { float* p = ws + o; o += n; return p; };

    float* Eg = alloc(GG * 64);
    float* Pg = alloc(GG * 64);
    float* Rp = alloc(GG * 64);
    float* Tg = alloc(GG * 64);
    float* pos0 = alloc(GG * 64);
    float* posR = alloc(GG * 64);
    float* vb0 = alloc(GG * 64);
    float* vbr = alloc(GG * 64);
    float* z0 = alloc(GG * 64);
    float* pge2 = alloc(GG * 64);
    // ---- contiguous zero-init region
    float* zr0 = ws + o;
    float* aggP = alloc(GG * 64);
    float* aggS0 = alloc(GG * 64);
    float* aggS1 = alloc(GG * 64);
    float* statsE = alloc(128);
    float* S1 = alloc(128);
    float* S2 = alloc(128);
    float* S3 = alloc(256);
    float* S4 = alloc(128);
    int zcount = (int)((ws + o) - zr0);
    // ---- degrees init to 1.0 (contiguous)
    float* degC = alloc(GG);
    float* degS = alloc(GG);
    float* dinvC = alloc(GG);
    float* dc2 = alloc(GG);
    float* dinvS = alloc(GG);
    float* ds2 = alloc(GG);
    float* aE = alloc(64); float* cE = alloc(64);
    float* a1 = alloc(64); float* c1 = alloc(64);
    float* a2 = alloc(64); float* c2 = alloc(64);
    float* a3 = alloc(128); float* c3 = alloc(128);
    float* a4 = alloc(64); float* c4 = alloc(64);
    float* bump2 = alloc(BB * 64);
    float* h1pre = alloc((size_t)NN * 128);
    float* rpre = alloc((size_t)NN * 64);

    k_fill<<<512, 256, 0, stream>>>(zr0, 0.0f, zcount);
    k_fill<<<64, 256, 0, stream>>>(degC, 1.0f, 2 * GG);
    k_renorm<<<80, 256, 0, stream>>>(gene_emb, pert_emb, emb_pos, Eg, Rp, Pg, statsE);
    k_deg<<<512, 256, 0, stream>>>(ce, cw, se, sw, degC, degS);
    k_finalize<<<1, 64, 0, stream>>>(statsE, bnEg, bnEb, aE, cE, 64, (float)GG);
    k_gmisc<<<320, 256, 0, stream>>>(Eg, aE, cE, degC, degS, Tg, dinvC, dc2, dinvS, ds2);
    k_scatter<<<1024, 256, 0, stream>>>(ce, cw, dinvC, Pg, aggP, EE1);
    k_scatter<<<1024, 256, 0, stream>>>(se, sw, dinvS, Rp, aggS0, EE2);

    int gb = (((GG + 15) / 16) + 1) / 2;  // row-tile pairs for G-scale GEMMs
    k_gemm64<true, false><<<gb, 256, 0, stream>>>(aggP, Pg, dc2, posW, posb, pos0, GG);
    k_gemm64<false, false><<<gb, 256, 0, stream>>>(Pg, nullptr, nullptr, posW, posb,
                                                   posR, GG);
    k_gemm64<true, true><<<gb, 256, 0, stream>>>(aggS0, Rp, ds2, simW0, simb0, z0, GG);
    k_scatter<<<1024, 256, 0, stream>>>(se, sw, dinvS, z0, aggS1, EE2);
    k_gemm64<true, false><<<gb, 256, 0, stream>>>(aggS1, z0, ds2, simW1, simb1, pge2,
                                                  GG);

    k_vb<<<320, 256, 0, stream>>>(Tg, pos0, posR, vb0, vbr);
    k_bumpinit<<<8, 256, 0, stream>>>(pb, bump2);
    k_bump<<<640, 256, 0, stream>>>(x, pge2, bump2);

    k_stats1<<<640, 256, 0, stream>>>(vb0, vbr, bump2, x, pW, S1);
    k_finalize<<<1, 64, 0, stream>>>(S1, bn1g, bn1b, a1, c1, 64, (float)NN);
    k_stats2<<<640, 256, 0, stream>>>(vb0, vbr, bump2, x, pW, a1, c1, S2);
    k_finalize<<<1, 64, 0, stream>>>(S2, bn2g, bn2b, a2, c2, 64, (float)NN);

    k_gemm1<<<640, 256, 0, stream>>>(vb0, vbr, bump2, x, pW, a1, c1, a2, c2, rW1, rb1,
                                     h1pre, S3);
    k_finalize<<<1, 128, 0, stream>>>(S3, rbn1g, rbn1b, a3, c3, 128, (float)NN);
    k_gemm2<<<640, 256, 0, stream>>>(h1pre, a3, c3, rW2, rb2, rpre, S4);
    k_finalize<<<1, 64, 0, stream>>>(S4, rbn2g, rbn2b, a4, c4, 64, (float)NN);
    k_out<<<1024, 256, 0, stream>>>(rpre, a4, c4, iw, ib, x, out);
}